// ProposalGenerator_64785286693436
// MI455X (gfx1250) — compile-verified
//
#include <hip/hip_runtime.h>
#include <stdint.h>

#define N_PRE       6000
#define N_POST      1000
#define NMS_T       0.7f
#define MIN_SZ      16.0f
#define NBINS       65536
#define CAND_CAP    16384
#define NWORDS      188          // ceil(6000/32)
#define ROWS_PER_BLOCK 64

typedef float        f32x4 __attribute__((ext_vector_type(4)));
typedef unsigned int u32x4 __attribute__((ext_vector_type(4)));
typedef unsigned int u32x8 __attribute__((ext_vector_type(8)));

// ---------------------------------------------------------------- decode ----
__device__ __forceinline__ void decode_box(int i,
    const int* __restrict__ bidx, const int* __restrict__ isz,
    const float* __restrict__ anchors, const float* __restrict__ deltas,
    float& x1, float& y1, float& x2, float& y2, bool& valid)
{
    // streaming, read-once inputs: non-temporal (th:TH_LOAD_NT) b128 loads
    f32x4 a = __builtin_nontemporal_load((const f32x4*)anchors + i);
    f32x4 d = __builtin_nontemporal_load((const f32x4*)deltas + i);
    int b = __builtin_nontemporal_load(bidx + i);
    float h = (float)isz[2 * b + 0];
    float w = (float)isz[2 * b + 1];
    float aw = a.z - a.x, ah = a.w - a.y;
    float ax = a.x + 0.5f * aw, ay = a.y + 0.5f * ah;
    float px = ax + d.x * aw, py = ay + d.y * ah;
    float dw = fminf(fmaxf(d.z, -10.f), 10.f);
    float dh = fminf(fmaxf(d.w, -10.f), 10.f);
    float pw = aw * __expf(dw), ph = ah * __expf(dh);
    x1 = fminf(fmaxf(px - 0.5f * pw, 0.f), w - 1.f);
    y1 = fminf(fmaxf(py - 0.5f * ph, 0.f), h - 1.f);
    x2 = fminf(fmaxf(px + 0.5f * pw, 0.f), w - 1.f);
    y2 = fminf(fmaxf(py + 0.5f * ph, 0.f), h - 1.f);
    valid = (x2 - x1 >= MIN_SZ) && (y2 - y1 >= MIN_SZ);
}

// ------------------------------------------------------------ init ----------
__global__ void init_ws(unsigned* __restrict__ hist, unsigned* __restrict__ meta,
                        unsigned* __restrict__ cand32)
{
    int i = blockIdx.x * blockDim.x + threadIdx.x;
    if (i < NBINS)        hist[i]   = 0u;
    if (i < 16)           meta[i]   = 0u;
    if (i < CAND_CAP * 2) cand32[i] = 0u;   // zero tail so bulk LDS copy pads with 0
}

// --------------------------------------------- decode + score keys + hist ---
__global__ void decode_keys(const int* __restrict__ bidx, const int* __restrict__ isz,
                            const float* __restrict__ anchors,
                            const float* __restrict__ logits,
                            const float* __restrict__ deltas,
                            unsigned* __restrict__ keys, unsigned* __restrict__ hist, int N)
{
    int i = blockIdx.x * blockDim.x + threadIdx.x;
    if (i >= N) return;
    float x1, y1, x2, y2; bool valid;
    decode_box(i, bidx, isz, anchors, deltas, x1, y1, x2, y2, valid);
    float l = __builtin_nontemporal_load(logits + i);
    float s = 1.f / (1.f + __expf(-l));          // (0,1): positive float bits order as u32
    unsigned key = valid ? __float_as_uint(s) : 0u;
    keys[i] = key;
    if (key) atomicAdd(&hist[key >> 16], 1u);    // invalids skipped: bin0 never reaches threshold
}

// ------------------------------------------------------- threshold select ---
__global__ __launch_bounds__(1024)
void find_thresh(const unsigned* __restrict__ hist, unsigned* __restrict__ meta)
{
    __shared__ unsigned csum[1024];
    int t = threadIdx.x;
    unsigned s = 0;
    for (int b = t * 64; b < t * 64 + 64; ++b) s += hist[b];
    csum[t] = s;
    __syncthreads();
    if (t == 0) {
        unsigned acc = 0;
        int chunk = 1023;
        for (; chunk >= 0; --chunk) {
            if (acc + csum[chunk] >= N_PRE) break;
            acc += csum[chunk];
        }
        unsigned above = acc;
        int B = 0;
        if (chunk >= 0) {
            int b = chunk * 64 + 63;
            for (; b > chunk * 64; --b) {
                if (above + hist[b] >= N_PRE) break;
                above += hist[b];
            }
            B = b;
        }
        meta[0] = (unsigned)B;   // threshold bin
        meta[1] = above;         // count strictly above bin B
    }
}

// -------------------------------------------------------------- compact -----
__global__ void compact_cand(const unsigned* __restrict__ keys, unsigned* __restrict__ meta,
                             unsigned long long* __restrict__ cand, int N)
{
    int i = blockIdx.x * blockDim.x + threadIdx.x;
    if (i >= N) return;
    unsigned key = __builtin_nontemporal_load(keys + i);   // read-once stream
    unsigned B = meta[0];
    if (key && (key >> 16) >= B) {
        unsigned pos = atomicAdd(&meta[2], 1u);
        if (pos < CAND_CAP)
            cand[pos] = ((unsigned long long)key << 32) | (unsigned)(~(unsigned)i); // ~i: ties -> lower idx
    }
}

// ---- single-WG bitonic sort (128 KB LDS), staged via async global->LDS -----
__global__ __launch_bounds__(1024)
void sort_select(const unsigned long long* __restrict__ cand, const unsigned* __restrict__ meta,
                 unsigned* __restrict__ topkey, unsigned* __restrict__ topidx)
{
    __shared__ unsigned long long sdat[CAND_CAP];   // 128 KB of CDNA5's 320 KB LDS/WG
    // ASYNCcnt-tracked DMA: 16 B per lane per op, tail already zeroed by init_ws
    for (int p = threadIdx.x * 2; p < CAND_CAP; p += 1024 * 2) {
        unsigned lds_off = (unsigned)(unsigned long long)(uintptr_t)(&sdat[p]);
        unsigned long long gaddr = (unsigned long long)(uintptr_t)(cand + p);
        asm volatile("global_load_async_to_lds_b128 %0, %1, off"
                     :: "v"(lds_off), "v"(gaddr) : "memory");
    }
    asm volatile("s_wait_asynccnt 0x0" ::: "memory");
    __syncthreads();
    for (int k = 2; k <= CAND_CAP; k <<= 1) {
        for (int j = k >> 1; j > 0; j >>= 1) {
            for (int idx = threadIdx.x; idx < CAND_CAP; idx += 1024) {
                int ixj = idx ^ j;
                if (ixj > idx) {
                    unsigned long long a = sdat[idx], b = sdat[ixj];
                    bool asc = (idx & k) != 0;               // descending overall
                    if (asc ? (a > b) : (a < b)) { sdat[idx] = b; sdat[ixj] = a; }
                }
            }
            __syncthreads();
        }
    }
    (void)meta;
    for (int p = threadIdx.x; p < N_PRE; p += 1024) {
        unsigned long long v = sdat[p];
        topkey[p] = (unsigned)(v >> 32);
        topidx[p] = ~((unsigned)v);
    }
}

// --------------------------- re-decode only the 6000 winners ----------------
__global__ void gather_decode(const unsigned* __restrict__ topkey, const unsigned* __restrict__ topidx,
                              const int* __restrict__ bidx, const int* __restrict__ isz,
                              const float* __restrict__ anchors, const float* __restrict__ deltas,
                              float4* __restrict__ boxes, int* __restrict__ tbidx, int N)
{
    int j = blockIdx.x * blockDim.x + threadIdx.x;
    if (j >= N_PRE) return;
    unsigned key = topkey[j];
    unsigned idx = topidx[j];
    if (key != 0u && idx < (unsigned)N) {
        float x1, y1, x2, y2; bool valid;
        decode_box((int)idx, bidx, isz, anchors, deltas, x1, y1, x2, y2, valid);
        boxes[j] = make_float4(x1, y1, x2, y2);
        tbidx[j] = bidx[idx];
    } else {
        boxes[j] = make_float4(0.f, 0.f, 0.f, 0.f);
        tbidx[j] = -1;
    }
}

// ------ NMS bitmask: TDM (tensor DMA) LDS staging + wave32 ballot -----------
__global__ __launch_bounds__(256)
void nms_mask(const float4* __restrict__ boxes, unsigned* __restrict__ mask)
{
    __shared__ float4 sb[N_PRE];                  // 96 KB of the 320 KB LDS
    // One Tensor Data Mover descriptor copies the whole 96 KB table in one op.
    // D# group0: count=1 | lds_addr | global_addr[56:0] | type=2
    // D# group1: data_size=8B, tensor/tile dim0 = 12000 units, 1 row
    if (threadIdx.x < 32) {                       // wave 0 only (EXEC ignored by TDM)
        unsigned long long g = (unsigned long long)(uintptr_t)boxes;
        unsigned lds_base = (unsigned)(unsigned long long)(uintptr_t)(&sb[0]);
        const unsigned UNITS = (N_PRE * 16u) / 8u;          // 12000 x 8B = 96000 B
        u32x4 g0;
        g0[0] = 1u;                                         // count=1, user mode
        g0[1] = lds_base;                                   // lds_addr
        g0[2] = (unsigned)(g & 0xFFFFFFFFu);                // global_addr[31:0]
        g0[3] = (unsigned)((g >> 32) & 0x01FFFFFFu) | (2u << 30); // addr[56:32] | type=2
        u32x8 g1;
        g1[0] = 3u << 16;                                   // data_size=3 (8B), mask=0
        g1[1] = (UNITS & 0xFFFFu) << 16;                    // tensor_dim0[15:0]
        g1[2] = (UNITS >> 16) | (1u << 16);                 // tensor_dim0[31:16], tensor_dim1=1
        g1[3] = (UNITS & 0xFFFFu) << 16;                    // tile_dim0
        g1[4] = 1u;                                         // tile_dim1=1, tile_dim2=0
        g1[5] = UNITS;                                      // tensor_dim0_stride[31:0]
        g1[6] = 0u;
        g1[7] = 0u;
        asm volatile("tensor_load_to_lds %0, %1" :: "s"(g0), "s"(g1) : "memory");
        asm volatile("s_wait_tensorcnt 0x0" ::: "memory");
    }
    __syncthreads();

    int wave = threadIdx.x >> 5;                  // 8 waves / block
    int lane = threadIdx.x & 31;
    int rowBase = blockIdx.x * ROWS_PER_BLOCK;
    for (int r = wave; r < ROWS_PER_BLOCK; r += 8) {
        int i = rowBase + r;
        if (i >= N_PRE) break;                    // uniform per wave
        float4 bi = sb[i];
        float areai = (bi.z - bi.x) * (bi.w - bi.y);
        for (int w = 0; w < NWORDS; ++w) {
            int j = w * 32 + lane;
            bool sup = false;
            if (j < N_PRE && j > i) {
                float4 bj = sb[j];
                float xx1 = fmaxf(bi.x, bj.x), yy1 = fmaxf(bi.y, bj.y);
                float xx2 = fminf(bi.z, bj.z), yy2 = fminf(bi.w, bj.w);
                float inter = fmaxf(xx2 - xx1, 0.f) * fmaxf(yy2 - yy1, 0.f);
                float areaj = (bj.z - bj.x) * (bj.w - bj.y);
                float iou = inter / (areai + areaj - inter + 1e-9f);
                sup = iou > NMS_T;
            }
            unsigned word = __builtin_amdgcn_ballot_w32(sup);
            if (lane == 0) mask[(size_t)i * NWORDS + w] = word;
        }
    }
}

// ------------- serial greedy scan: ONE wave32, prefetched mask rows ---------
__global__ __launch_bounds__(32)
void nms_scan(const unsigned* __restrict__ mask, const unsigned* __restrict__ topkey,
              unsigned* __restrict__ keepIdx)
{
    __shared__ unsigned ssup[192];                // 188 words, lane owns w = lane + 32q
    int lane = threadIdx.x;
    for (int w = lane; w < 192; w += 32) ssup[w] = 0u;
    __syncthreads();                              // NOP: single-wave workgroup

    unsigned cur[6], nxt[6];
#pragma unroll
    for (int q = 0; q < 6; ++q) {
        int w = lane + 32 * q;
        cur[q] = (w < NWORDS) ? mask[w] : 0u;
    }
    for (int i = 0; i < N_PRE; ++i) {
        // software-prefetch next mask row while deciding on this one
#pragma unroll
        for (int q = 0; q < 6; ++q) {
            int w = lane + 32 * q;
            nxt[q] = (i + 1 < N_PRE && w < NWORDS) ? mask[(size_t)(i + 1) * NWORDS + w] : 0u;
        }
        unsigned supw = ssup[i >> 5];             // LDS broadcast of owner lane's word
        if (!((supw >> (i & 31)) & 1u)) {
#pragma unroll
            for (int q = 0; q < 6; ++q) {
                int w = lane + 32 * q;
                if (w < NWORDS) ssup[w] |= cur[q];
            }
        }
        __syncthreads();                          // NOP barrier, orders LDS for compiler
#pragma unroll
        for (int q = 0; q < 6; ++q) cur[q] = nxt[q];
    }

    // compact first N_POST kept indices via wave32 ballot prefix
    int total = 0;
    for (int base = 0; base < N_PRE; base += 32) {
        int i = base + lane;
        bool kept = false;
        if (i < N_PRE)
            kept = !((ssup[i >> 5] >> (i & 31)) & 1u) && (topkey[i] != 0u);
        unsigned bal = __builtin_amdgcn_ballot_w32(kept);
        int pos = total + __popc(bal & ((1u << lane) - 1u));
        if (kept && pos < N_POST) keepIdx[pos] = (unsigned)i;
        total += __popc(bal);
        if (total >= N_POST) break;
    }
    int start = total < N_POST ? total : N_POST;
    for (int j = start + lane; j < N_POST; j += 32) keepIdx[j] = N_PRE;  // sentinel pad row
}

// ---------------------------------------------------------------- output ----
__global__ void write_out(const unsigned* __restrict__ keepIdx, const float4* __restrict__ boxes,
                          const int* __restrict__ tbidx, float* __restrict__ out)
{
    int j = blockIdx.x * blockDim.x + threadIdx.x;
    if (j >= N_POST) return;
    unsigned k = keepIdx[j];
    float4 b = make_float4(0.f, 0.f, 0.f, 0.f);
    int bi = -1;
    if (k < N_PRE) { b = boxes[k]; bi = tbidx[k]; }
    out[j * 4 + 0] = b.x;
    out[j * 4 + 1] = b.y;
    out[j * 4 + 2] = b.z;
    out[j * 4 + 3] = b.w;
    ((int*)out)[N_POST * 4 + j] = bi;             // int32 batch indices after 1000x4 floats
}

// ---------------------------------------------------------------- launch ----
extern "C" void kernel_launch(void* const* d_in, const int* in_sizes, int n_in,
                              void* d_out, int out_size, void* d_ws, size_t ws_size,
                              hipStream_t stream)
{
    const int*   bidx    = (const int*)d_in[0];
    const int*   isz     = (const int*)d_in[1];
    const float* anchors = (const float*)d_in[2];
    const float* logits  = (const float*)d_in[3];
    const float* deltas  = (const float*)d_in[4];
    const int N = in_sizes[0];

    char* ws = (char*)d_ws;
    size_t off = 0;
    auto take = [&](size_t bytes) -> void* {
        void* p = ws + off;
        off = (off + bytes + 255) & ~(size_t)255;
        return p;
    };
    unsigned*            keys   = (unsigned*)take((size_t)N * 4);
    unsigned*            hist   = (unsigned*)take((size_t)NBINS * 4);
    unsigned*            meta   = (unsigned*)take(64);
    unsigned long long*  cand   = (unsigned long long*)take((size_t)CAND_CAP * 8);
    unsigned*            topkey = (unsigned*)take((size_t)N_PRE * 4);
    unsigned*            topidx = (unsigned*)take((size_t)N_PRE * 4);
    float4*              boxes  = (float4*)take((size_t)N_PRE * 16);
    int*                 tbidx  = (int*)take((size_t)N_PRE * 4);
    unsigned*            mask   = (unsigned*)take((size_t)N_PRE * NWORDS * 4);
    unsigned*            keepIdx= (unsigned*)take((size_t)N_POST * 4);

    const int blk = 256;
    hipLaunchKernelGGL(init_ws,      dim3(256), dim3(256), 0, stream, hist, meta, (unsigned*)cand);
    hipLaunchKernelGGL(decode_keys,  dim3((N + blk - 1) / blk), dim3(blk), 0, stream,
                       bidx, isz, anchors, logits, deltas, keys, hist, N);
    hipLaunchKernelGGL(find_thresh,  dim3(1), dim3(1024), 0, stream, hist, meta);
    hipLaunchKernelGGL(compact_cand, dim3((N + blk - 1) / blk), dim3(blk), 0, stream,
                       keys, meta, cand, N);
    hipLaunchKernelGGL(sort_select,  dim3(1), dim3(1024), 0, stream, cand, meta, topkey, topidx);
    hipLaunchKernelGGL(gather_decode,dim3((N_PRE + blk - 1) / blk), dim3(blk), 0, stream,
                       topkey, topidx, bidx, isz, anchors, deltas, boxes, tbidx, N);
    hipLaunchKernelGGL(nms_mask,     dim3((N_PRE + ROWS_PER_BLOCK - 1) / ROWS_PER_BLOCK),
                       dim3(256), 0, stream, boxes, mask);
    hipLaunchKernelGGL(nms_scan,     dim3(1), dim3(32), 0, stream, mask, topkey, keepIdx);
    hipLaunchKernelGGL(write_out,    dim3((N_POST + 255) / 256), dim3(256), 0, stream,
                       keepIdx, boxes, tbidx, (float*)d_out);
    (void)n_in; (void)out_size; (void)ws_size;
}